// PointToClusterModule_61349312856622
// MI455X (gfx1250) — compile-verified
//
#include <hip/hip_runtime.h>
#include <math.h>

// Problem constants (from reference): B=4, N=40000, F=4, C=128, K=256, SIZE=256
#define BATCH 4
#define NPTS  40000
#define FF    4
#define CC    128
#define KK    256
#define SZ    256

typedef __attribute__((ext_vector_type(2))) float v2f;
typedef __attribute__((ext_vector_type(8))) float v8f;

// ---------------------------------------------------------------------------
// Kernel 1: compute prob[b][k] = softmax_k( sum_c proposals[b][c][k] * vc[c] )
// where vc[c] = W1[F+c,:] . W2.   One block, 4 waves (one wave per batch).
// The sc GEMV runs on the tensor core via V_WMMA_F32_16X16X4_F32 (exact fp32).
// ---------------------------------------------------------------------------
__global__ __launch_bounds__(128) void score_softmax_kernel(
    const float* __restrict__ proposals,  // (B, C, K) row-major
    const float* __restrict__ W1,         // (F+C, SIZE)
    const float* __restrict__ W2,         // (SIZE, 1)
    float* __restrict__ prob)             // (B, K) output (workspace)
{
    __shared__ float vc[CC];
    __shared__ float sc[BATCH * KK];
    __shared__ float red[2 * BATCH];      // per-batch max, 1/sum

    const int t = threadIdx.x;            // 0..127

    // ---- vc[c] = W1[F+c,:] . W2  (thread t handles channel c = t) ----
    {
        const float* w1row = W1 + (size_t)(FF + t) * SZ;
        float acc = 0.0f;
        #pragma unroll 8
        for (int s = 0; s < SZ; ++s) acc = fmaf(w1row[s], W2[s], acc);
        vc[t] = acc;
    }
    __syncthreads();

    // ---- sc[b][k] via WMMA: D(16x16) = A(16x4) x B(4x16) accumulated over c ----
    // wave w owns batch b = w. A[m][j] = proposals[b][c0+j][k0+m] (m = k-row),
    // B[j][n] = vc[c0+j] replicated over all n columns -> every column of D
    // equals sc[k0 + m].
    const int wave = t >> 5;              // batch index 0..3
    const int lane = t & 31;
    const int m    = lane & 15;           // k-row within tile
    const int jb   = (lane >> 4) << 1;    // K-pair base: 0 (lanes 0-15) or 2 (lanes 16-31)
    const float* Pb = proposals + (size_t)wave * CC * KK;

    for (int k0 = 0; k0 < KK; k0 += 16) {
        v8f acc = {0.f, 0.f, 0.f, 0.f, 0.f, 0.f, 0.f, 0.f};
        #pragma unroll 4
        for (int c0 = 0; c0 < CC; c0 += 4) {
            // A fragment (16x4 f32, 2 VGPRs): VGPR0 = K{0,2}, VGPR1 = K{1,3}
            v2f a, bf;
            a.x = Pb[(size_t)(c0 + jb)     * KK + k0 + m];
            a.y = Pb[(size_t)(c0 + jb + 1) * KK + k0 + m];
            // B fragment (4x16 f32, 2 VGPRs): rows {0,2}/{1,3} split at lane 16,
            // value replicated across the 16 columns.
            bf.x = vc[c0 + jb];
            bf.y = vc[c0 + jb + 1];
            acc = __builtin_amdgcn_wmma_f32_16x16x4_f32(
                /*neg_a=*/false, a, /*neg_b=*/false, bf,
                /*c_mod=*/(short)0, acc, /*reuse_a=*/false, /*reuse_b=*/false);
        }
        // D layout: VGPR r holds row r (lanes 0-15) and row 8+r (lanes 16-31);
        // all columns identical, so lane 0 / lane 16 carry the full tile.
        if (lane == 0) {
            #pragma unroll
            for (int r = 0; r < 8; ++r) sc[wave * KK + k0 + r] = acc[r];
        } else if (lane == 16) {
            #pragma unroll
            for (int r = 0; r < 8; ++r) sc[wave * KK + k0 + 8 + r] = acc[r];
        }
    }
    __syncthreads();

    // ---- softmax statistics per batch (threads 0..3) ----
    if (t < BATCH) {
        float mx = -3.402823466e38f;
        for (int k = 0; k < KK; ++k) mx = fmaxf(mx, sc[t * KK + k]);
        float sum = 0.0f;
        for (int k = 0; k < KK; ++k) sum += __expf(sc[t * KK + k] - mx);
        red[t]         = mx;
        red[BATCH + t] = 1.0f / sum;
    }
    __syncthreads();

    // ---- normalized probabilities to workspace ----
    for (int i = t; i < BATCH * KK; i += 128) {
        int b = i >> 8;                   // KK == 256
        prob[i] = __expf(sc[i] - red[b]) * red[BATCH + b];
    }
}

// ---------------------------------------------------------------------------
// Kernel 2: broadcast prob[b][:] into out[b][n][:] for all n.
// Pure bandwidth: 164 MB of b128 stores (~7 us at 23.3 TB/s).
// 256 threads/block = 4 rows of 256 floats; 40000 blocks cover B*N rows.
// Batch boundaries (multiples of 40000 rows) align with 4-row blocks.
// ---------------------------------------------------------------------------
__global__ __launch_bounds__(256) void broadcast_kernel(
    const float4* __restrict__ prob4,     // (B, 64) float4
    float4* __restrict__ out4)            // (B*N, 64) float4
{
    const int b = blockIdx.x / (NPTS / 4);            // 10000 blocks per batch
    const int t = threadIdx.x;
    const float4 v = prob4[b * (KK / 4) + (t & 63)];  // hot 4 KB table (L2/WGP$)
    out4[(size_t)blockIdx.x * 256 + t] = v;
}

extern "C" void kernel_launch(void* const* d_in, const int* in_sizes, int n_in,
                              void* d_out, int out_size, void* d_ws, size_t ws_size,
                              hipStream_t stream) {
    // Inputs (setup_inputs order): point_clouds, proposals, W1, b1, W2, b2.
    // point_clouds, b1, b2 provably cancel under softmax shift-invariance.
    const float* proposals = (const float*)d_in[1];
    const float* W1        = (const float*)d_in[2];
    const float* W2        = (const float*)d_in[4];
    float*       prob      = (float*)d_ws;            // 4 KB: (B, K) probabilities
    float*       out       = (float*)d_out;           // (B, N, K) fp32

    score_softmax_kernel<<<1, 128, 0, stream>>>(proposals, W1, W2, prob);
    broadcast_kernel<<<(BATCH * NPTS) / 4, 256, 0, stream>>>(
        (const float4*)prob, (float4*)out);
    (void)in_sizes; (void)n_in; (void)out_size; (void)ws_size; (void)out;
}